// GNNModel_5600637354562
// MI455X (gfx1250) — compile-verified
//
#include <hip/hip_runtime.h>
#include <hip/hip_bf16.h>

typedef float v2f __attribute__((ext_vector_type(2)));
typedef float v8f __attribute__((ext_vector_type(8)));

#define N_NODES 50000
#define N_EDGES 800000
#define IN_DIM  256
#define HID_DIM 128
#define OUT_DIM 64
#define BN_EPS  1e-5f

// ---------------------------------------------------------------------------
// Degree / norm kernels
// ---------------------------------------------------------------------------
__global__ void deg_init_kernel(float* __restrict__ deg, int n) {
    int t = blockIdx.x * blockDim.x + threadIdx.x;
    if (t < n) deg[t] = 1.0f;   // self loop contributes 1
}

__global__ void deg_acc_kernel(const int* __restrict__ dst, float* __restrict__ deg, int E) {
    int t = blockIdx.x * blockDim.x + threadIdx.x;
    if (t < E) atomicAdd(&deg[dst[t]], 1.0f);
}

__global__ void deg_to_dinv_kernel(float* __restrict__ deg, int n) {
    int t = blockIdx.x * blockDim.x + threadIdx.x;
    if (t < n) deg[t] = rsqrtf(deg[t]);   // deg >= 1 always (self loops)
}

// ---------------------------------------------------------------------------
// Weight repack: Wp[(k/2)*F + n] = float2{ W[k][n], W[k+1][n] }
// Turns the per-lane B-fragment (two rows strided by F) into one b64 load.
// ---------------------------------------------------------------------------
__global__ void repack_w_kernel(const float* __restrict__ W, float* __restrict__ Wp,
                                int half, int F) {
    int t = blockIdx.x * blockDim.x + threadIdx.x;
    if (t >= half * F) return;
    int th = t / F;
    int n  = t - th * F;
    float2 v;
    v.x = W[(size_t)(2 * th)     * F + n];
    v.y = W[(size_t)(2 * th + 1) * F + n];
    *(float2*)(Wp + (size_t)t * 2) = v;
}

// ---------------------------------------------------------------------------
// WMMA FP32 GEMM: H[nRows x F] = X[nRows x K] @ W[K x F]  (W pre-packed)
// One wave32 per 16-row stripe, all F/16 column tiles accumulated in regs.
// A frag (16x4 f32): lane holds X[row0 + lane%16][k + 2*(lane/16) + {0,1}]
// B frag (4x16 f32): lane holds Wp pair at (k/2 + lane/16, ct*16 + lane%16)
// C/D (16x16 f32):  vgpr j, lanes0-15 -> M=j, lanes16-31 -> M=j+8; N=lane%16
// Double-buffered: next k-step fragments fetched before current WMMA batch.
// ---------------------------------------------------------------------------
template<int K, int F>
__global__ __launch_bounds__(256) void gemm_wmma_kernel(const float* __restrict__ X,
                                                        const float* __restrict__ Wp,
                                                        float* __restrict__ H,
                                                        int nRows) {
    constexpr int CT = F / 16;
    const int rowTile = blockIdx.x * (blockDim.x >> 5) + (threadIdx.x >> 5);
    if (rowTile >= (nRows >> 4)) return;     // wave-uniform: EXEC stays all-ones

    const int lane = threadIdx.x & 31;
    const int m  = lane & 15;                // row within tile / N column index
    const int hi = lane >> 4;                // half-wave select
    const int kh = hi << 1;                  // K sub-offset: 0 or 2
    const int row0 = rowTile << 4;

    const float* __restrict__ xrow  = X + (size_t)(row0 + m) * K;
    const float* __restrict__ wbase = Wp + ((size_t)hi * F + m) * 2;  // pair (hi, m)

    v8f acc[CT];
#pragma unroll
    for (int c = 0; c < CT; ++c) acc[c] = 0.0f;

    // prologue: fragments for k = 0
    v2f a = *(const v2f*)(xrow + kh);
    v2f b[CT];
#pragma unroll
    for (int c = 0; c < CT; ++c) b[c] = *(const v2f*)(wbase + c * 32);

#pragma unroll 4
    for (int k = 0; k < K; k += 4) {
        v2f an = a;
        v2f bn[CT];
#pragma unroll
        for (int c = 0; c < CT; ++c) bn[c] = b[c];
        const int kn = k + 4;
        if (kn < K) {                         // wave-uniform branch
            __builtin_prefetch(xrow + kn + 32, 0, 1);   // global_prefetch_b8
            an = *(const v2f*)(xrow + kn + kh);
            const float* wn = wbase + (size_t)(kn >> 1) * (2 * F);
#pragma unroll
            for (int c = 0; c < CT; ++c) bn[c] = *(const v2f*)(wn + c * 32);
        }
#pragma unroll
        for (int c = 0; c < CT; ++c) {
            acc[c] = __builtin_amdgcn_wmma_f32_16x16x4_f32(
                false, a, false, b[c], (short)0, acc[c], false, false);
        }
        a = an;
#pragma unroll
        for (int c = 0; c < CT; ++c) b[c] = bn[c];
    }

    const int rbase = row0 + (hi << 3);       // +8 rows for upper half-lanes
#pragma unroll
    for (int c = 0; c < CT; ++c) {
#pragma unroll
        for (int j = 0; j < 8; ++j) {
            H[(size_t)(rbase + j) * F + c * 16 + m] = acc[c][j];
        }
    }
}

// ---------------------------------------------------------------------------
// Aggregation init: agg[i] = dinv[i]^2 * h[i] + bias   (self-loop + bias term)
// ---------------------------------------------------------------------------
template<int F>
__global__ void agg_init_kernel(const float* __restrict__ h,
                                const float* __restrict__ dinv,
                                const float* __restrict__ bias,
                                float* __restrict__ agg, int n) {
    constexpr int CPN = F / 4;
    int t = blockIdx.x * blockDim.x + threadIdx.x;
    if (t >= n * CPN) return;
    int i = t / CPN;
    int c = (t - i * CPN) * 4;
    float w = dinv[i]; w *= w;
    const float4 hv = *(const float4*)(h + (size_t)i * F + c);
    float4 o;
    o.x = fmaf(hv.x, w, bias[c + 0]);
    o.y = fmaf(hv.y, w, bias[c + 1]);
    o.z = fmaf(hv.z, w, bias[c + 2]);
    o.w = fmaf(hv.w, w, bias[c + 3]);
    *(float4*)(agg + (size_t)i * F + c) = o;
}

// ---------------------------------------------------------------------------
// Edge scatter: agg[dst] += dinv[src]*dinv[dst] * h[src]
// F=128: one wave per edge (32 lanes x float4). Atomics resolve in L2.
// ---------------------------------------------------------------------------
template<int F>
__global__ void agg_edges_kernel(const float* __restrict__ h,
                                 const float* __restrict__ dinv,
                                 const int* __restrict__ src,
                                 const int* __restrict__ dst,
                                 float* __restrict__ agg, int E) {
    constexpr int CPE = F / 4;
    int t = blockIdx.x * blockDim.x + threadIdx.x;
    int e = t / CPE;
    if (e >= E) return;
    int c = (t - e * CPE) * 4;
    int s = src[e];
    int d = dst[e];
    float w = dinv[s] * dinv[d];
    const float4 hv = *(const float4*)(h + (size_t)s * F + c);
    float* ap = agg + (size_t)d * F + c;
    atomicAdd(ap + 0, hv.x * w);
    atomicAdd(ap + 1, hv.y * w);
    atomicAdd(ap + 2, hv.z * w);
    atomicAdd(ap + 3, hv.w * w);
}

// ---------------------------------------------------------------------------
// BatchNorm (eval) + ReLU, in place
// ---------------------------------------------------------------------------
__global__ void bn_relu_kernel(float* __restrict__ h,
                               const float* __restrict__ gamma,
                               const float* __restrict__ beta,
                               const float* __restrict__ mean,
                               const float* __restrict__ var,
                               int total, int F) {
    int t = blockIdx.x * blockDim.x + threadIdx.x;
    if (t >= total) return;
    int f = t % F;
    float scale = gamma[f] * rsqrtf(var[f] + BN_EPS);
    float v = (h[t] - mean[f]) * scale + beta[f];
    h[t] = fmaxf(v, 0.0f);
}

// ---------------------------------------------------------------------------
// Host-side orchestration
// ---------------------------------------------------------------------------
static inline int cdiv(long long a, long long b) { return (int)((a + b - 1) / b); }

extern "C" void kernel_launch(void* const* d_in, const int* in_sizes, int n_in,
                              void* d_out, int out_size, void* d_ws, size_t ws_size,
                              hipStream_t stream) {
    const float* x      = (const float*)d_in[0];
    const int*   ei     = (const int*)d_in[1];
    const float* W1     = (const float*)d_in[2];
    const float* b1     = (const float*)d_in[3];
    const float* W2     = (const float*)d_in[4];
    const float* b2     = (const float*)d_in[5];
    const float* W3     = (const float*)d_in[6];
    const float* b3     = (const float*)d_in[7];
    const float* gamma1 = (const float*)d_in[8];
    const float* beta1  = (const float*)d_in[9];
    const float* mean1  = (const float*)d_in[10];
    const float* var1   = (const float*)d_in[11];
    const float* gamma2 = (const float*)d_in[12];
    const float* beta2  = (const float*)d_in[13];
    const float* mean2  = (const float*)d_in[14];
    const float* var2   = (const float*)d_in[15];

    const int n = in_sizes[0] / IN_DIM;     // 50000
    const int E = in_sizes[1] / 2;          // 800000
    const int* src = ei;
    const int* dst = ei + E;

    float* out = (float*)d_out;

    // Workspace carve-up (all regions written before read)
    char* ws = (char*)d_ws;
    float* dinv = (float*)ws;                                     // n floats
    float* A    = (float*)(ws + (1u << 20));                      // n x 128
    float* B    = (float*)(ws + (1u << 20) + 26u * 1024 * 1024);  // n x 128
    float* Wp1  = (float*)(ws + 54u * 1024 * 1024);               // 128 KB
    float* Wp2  = (float*)(ws + 54u * 1024 * 1024 + 256u * 1024); //  64 KB
    float* Wp3  = (float*)(ws + 54u * 1024 * 1024 + 512u * 1024); //  32 KB

    const int TB = 256;

    // --- degree / symmetric norm + weight repacks ---
    deg_init_kernel<<<cdiv(n, TB), TB, 0, stream>>>(dinv, n);
    deg_acc_kernel<<<cdiv(E, TB), TB, 0, stream>>>(dst, dinv, E);
    deg_to_dinv_kernel<<<cdiv(n, TB), TB, 0, stream>>>(dinv, n);
    repack_w_kernel<<<cdiv((IN_DIM / 2) * HID_DIM, TB), TB, 0, stream>>>(W1, Wp1, IN_DIM / 2, HID_DIM);
    repack_w_kernel<<<cdiv((HID_DIM / 2) * HID_DIM, TB), TB, 0, stream>>>(W2, Wp2, HID_DIM / 2, HID_DIM);
    repack_w_kernel<<<cdiv((HID_DIM / 2) * OUT_DIM, TB), TB, 0, stream>>>(W3, Wp3, HID_DIM / 2, OUT_DIM);

    const int nTiles = n / 16;                      // 3125
    const int gemmBlocks = cdiv(nTiles, TB / 32);   // 8 waves per block

    // --- layer 1: GCNConv(256->128) + BN + ReLU ---
    gemm_wmma_kernel<IN_DIM, HID_DIM><<<gemmBlocks, TB, 0, stream>>>(x, Wp1, A, n);
    agg_init_kernel<HID_DIM><<<cdiv((long long)n * (HID_DIM / 4), TB), TB, 0, stream>>>(A, dinv, b1, B, n);
    agg_edges_kernel<HID_DIM><<<cdiv((long long)E * (HID_DIM / 4), TB), TB, 0, stream>>>(A, dinv, src, dst, B, E);
    bn_relu_kernel<<<cdiv((long long)n * HID_DIM, TB), TB, 0, stream>>>(B, gamma1, beta1, mean1, var1, n * HID_DIM, HID_DIM);

    // --- layer 2: GCNConv(128->128) + BN + ReLU ---
    gemm_wmma_kernel<HID_DIM, HID_DIM><<<gemmBlocks, TB, 0, stream>>>(B, Wp2, A, n);
    agg_init_kernel<HID_DIM><<<cdiv((long long)n * (HID_DIM / 4), TB), TB, 0, stream>>>(A, dinv, b2, B, n);
    agg_edges_kernel<HID_DIM><<<cdiv((long long)E * (HID_DIM / 4), TB), TB, 0, stream>>>(A, dinv, src, dst, B, E);
    bn_relu_kernel<<<cdiv((long long)n * HID_DIM, TB), TB, 0, stream>>>(B, gamma2, beta2, mean2, var2, n * HID_DIM, HID_DIM);

    // --- layer 3: GCNConv(128->64), no BN/ReLU, writes d_out ---
    gemm_wmma_kernel<HID_DIM, OUT_DIM><<<gemmBlocks, TB, 0, stream>>>(B, Wp3, A, n);
    agg_init_kernel<OUT_DIM><<<cdiv((long long)n * (OUT_DIM / 4), TB), TB, 0, stream>>>(A, dinv, b3, out, n);
    agg_edges_kernel<OUT_DIM><<<cdiv((long long)E * (OUT_DIM / 4), TB), TB, 0, stream>>>(A, dinv, src, dst, out, E);
}